// WHAI_90683939488225
// MI455X (gfx1250) — compile-verified
//
#include <hip/hip_runtime.h>
#include <math.h>

#define EPSC 2.2e-10f

typedef __attribute__((ext_vector_type(2))) float v2f;
typedef __attribute__((ext_vector_type(4))) float v4f;
typedef __attribute__((ext_vector_type(8))) float v8f;
typedef __attribute__((ext_vector_type(4))) int v4i;

#if defined(__AMDGCN__) &&                                                     \
    __has_builtin(__builtin_amdgcn_global_load_async_to_lds_b128) &&           \
    __has_builtin(__builtin_amdgcn_s_wait_asynccnt)
#define USE_ASYNC 1
typedef __attribute__((address_space(1))) v4i* g4p;
typedef __attribute__((address_space(3))) v4i* l4p;
#else
#define USE_ASYNC 0
#endif

// 16B/lane global -> LDS copy on the ASYNCcnt DMA path (fallback: via VGPRs)
__device__ __forceinline__ void cp16_g2l(const float* gp, float* sp) {
#if USE_ASYNC
  __builtin_amdgcn_global_load_async_to_lds_b128((g4p)gp, (l4p)sp, 0, 0);
#else
  *(v4f*)sp = *(const v4f*)gp;
#endif
}

__device__ __forceinline__ void async_wait0() {
#if USE_ASYNC
  __builtin_amdgcn_s_wait_asynccnt(0);
#endif
}

__device__ __forceinline__ v8f wmma4(v2f a, v2f b, v8f c) {
  // V_WMMA_F32_16X16X4_F32 : D = A(16x4) x B(4x16) + C(16x16), all f32
  return __builtin_amdgcn_wmma_f32_16x16x4_f32(false, a, false, b, (short)0, c,
                                               false, false);
}

__device__ __forceinline__ float digammaf_dev(float x) {
  float r = 0.0f;
  for (int i = 0; i < 6; ++i) {
    if (x < 6.0f) { r -= 1.0f / x; x += 1.0f; }
  }
  float inv = 1.0f / x;
  float inv2 = inv * inv;
  r += logf(x) - 0.5f * inv
     - inv2 * (1.0f / 12.0f - inv2 * (1.0f / 120.0f - inv2 * (1.0f / 252.0f)));
  return r;
}

// ---------------- transpose: x(B,V) -> xT(V,B) ----------------
__global__ __launch_bounds__(256) void transpose_k(const float* __restrict__ x,
                                                   float* __restrict__ xT,
                                                   int Bdim, int Vdim) {
  __shared__ float tile[32][33];
  int v0 = blockIdx.x * 32, b0 = blockIdx.y * 32;
  int tx = threadIdx.x & 31, ty = threadIdx.x >> 5;
  for (int r = 0; r < 32; r += 8) {
    int b = b0 + ty + r, v = v0 + tx;
    if (b < Bdim && v < Vdim) tile[ty + r][tx] = x[(size_t)b * Vdim + v];
  }
  __syncthreads();
  for (int r = 0; r < 32; r += 8) {
    int v = v0 + ty + r, b = b0 + tx;
    if (v < Vdim && b < Bdim) xT[(size_t)v * Bdim + b] = tile[tx][ty + r];
  }
}

// ---- NN: ratio = f( phi(M,K) @ theta(K,N) , xc(M,N) ); N=512, K<=128 ----
__global__ __launch_bounds__(256) void gemm_nn_ratio(
    const float* __restrict__ A, const float* __restrict__ Bm,
    const float* __restrict__ xc, float* __restrict__ ratio,
    int M, int N, int K, int mode) {
  __shared__ __align__(16) float sA[64 * 68];    // [m][k], Kc<=64
  __shared__ __align__(16) float sB[64 * 132];   // [k][n], 128 cols
  const int tid = threadIdx.x;
  const int half = (tid >> 4) & 1, lrow = tid & 15;
  const int wave = tid >> 5;
  const int m_sub = wave & 3, n_half = wave >> 2;
  const int mblk = blockIdx.x, nblk = blockIdx.y;
  v8f acc0 = {}, acc1 = {}, acc2 = {}, acc3 = {};
  const int m_l = m_sub * 16;
  for (int kc = 0; kc < K; kc += 64) {
    int kcs = K - kc; if (kcs > 64) kcs = 64;
    int nv4 = kcs >> 2;
    for (int idx = tid; idx < 64 * nv4; idx += 256) {
      int r = idx / nv4, c4 = (idx - r * nv4) << 2;
      int gr = mblk * 64 + r; if (gr > M - 1) gr = M - 1;
      cp16_g2l(&A[(size_t)gr * K + kc + c4], &sA[r * 68 + c4]);
    }
    for (int idx = tid; idx < kcs * 32; idx += 256) {
      int r = idx >> 5, c4 = (idx & 31) << 2;
      cp16_g2l(&Bm[(size_t)(kc + r) * N + nblk * 128 + c4], &sB[r * 132 + c4]);
    }
    async_wait0();
    __syncthreads();
    for (int kk = 0; kk < kcs; kk += 4) {
      v2f a;
      a.x = sA[(m_l + lrow) * 68 + kk + 2 * half];
      a.y = sA[(m_l + lrow) * 68 + kk + 2 * half + 1];
      const float* bp0 = &sB[(kk + 2 * half) * 132 + n_half * 64 + lrow];
      const float* bp1 = &sB[(kk + 1 + 2 * half) * 132 + n_half * 64 + lrow];
      v2f b;
      b.x = bp0[0];  b.y = bp1[0];  acc0 = wmma4(a, b, acc0);
      b.x = bp0[16]; b.y = bp1[16]; acc1 = wmma4(a, b, acc1);
      b.x = bp0[32]; b.y = bp1[32]; acc2 = wmma4(a, b, acc2);
      b.x = bp0[48]; b.y = bp1[48]; acc3 = wmma4(a, b, acc3);
    }
    __syncthreads();
  }
  int rowb = mblk * 64 + m_l;
  if (rowb < M) {  // M % 16 == 0: tile fully valid or fully out
#define EPI_NN(ACC, J)                                                        \
    {                                                                         \
      int colb = nblk * 128 + n_half * 64 + (J) * 16 + lrow;                  \
      for (int r = 0; r < 8; ++r) {                                           \
        int row = rowb + r + 8 * half;                                        \
        float d = fmaxf(ACC[r], EPSC);                                        \
        float xv = xc[(size_t)row * N + colb];                                \
        float val = (mode == 0) ? (xv / d)                                    \
                                : (digammaf_dev(d + xv) - digammaf_dev(d));   \
        ratio[(size_t)row * N + colb] = val;                                  \
      }                                                                       \
    }
    EPI_NN(acc0, 0) EPI_NN(acc1, 1) EPI_NN(acc2, 2) EPI_NN(acc3, 3)
#undef EPI_NN
  }
}

// ---- NT: W = phi(M,N) .* ( A(M,K) @ B(N,K)^T );  K=512, N<=128 ----
__global__ __launch_bounds__(256) void gemm_nt_w(
    const float* __restrict__ A, const float* __restrict__ Bm,
    const float* __restrict__ phi, float* __restrict__ W,
    int M, int N, int K) {
  __shared__ __align__(16) float sA[64 * 68];    // [m][k]
  __shared__ __align__(16) float sBn[64 * 68];   // [n][k] (un-transposed)
  const int tid = threadIdx.x;
  const int half = (tid >> 4) & 1, lrow = tid & 15;
  const int wave = tid >> 5;
  const int m_sub = wave & 3, n_pair = wave >> 2;  // n tiles: n_pair*2 + {0,1}
  const int mblk = blockIdx.x, nblk = blockIdx.y;
  v8f acc0 = {}, acc1 = {};
  for (int kc = 0; kc < K; kc += 64) {
    for (int idx = tid; idx < 64 * 16; idx += 256) {
      int r = idx >> 4, c4 = (idx & 15) << 2;
      int gr = mblk * 64 + r; if (gr > M - 1) gr = M - 1;
      cp16_g2l(&A[(size_t)gr * K + kc + c4], &sA[r * 68 + c4]);
      int gn = nblk * 64 + r; if (gn > N - 1) gn = N - 1;
      cp16_g2l(&Bm[(size_t)gn * K + kc + c4], &sBn[r * 68 + c4]);
    }
    async_wait0();
    __syncthreads();
    for (int kk = 0; kk < 64; kk += 4) {
      v2f a;
      a.x = sA[(m_sub * 16 + lrow) * 68 + kk + 2 * half];
      a.y = sA[(m_sub * 16 + lrow) * 68 + kk + 2 * half + 1];
      const float* q0 = &sBn[(n_pair * 32 + lrow) * 68 + kk + 2 * half];
      const float* q1 = q0 + 16 * 68;
      v2f b;
      b.x = q0[0]; b.y = q0[1]; acc0 = wmma4(a, b, acc0);
      b.x = q1[0]; b.y = q1[1]; acc1 = wmma4(a, b, acc1);
    }
    __syncthreads();
  }
  int rowb = mblk * 64 + m_sub * 16;
  if (rowb < M) {
#define EPI_NT(ACC, JJ)                                                       \
    {                                                                         \
      int ctile = nblk * 64 + n_pair * 32 + (JJ) * 16;                        \
      if (ctile < N) {                                                        \
        int colb = ctile + lrow;                                              \
        for (int r = 0; r < 8; ++r) {                                         \
          int row = rowb + r + 8 * half;                                      \
          W[(size_t)row * N + colb] = phi[(size_t)row * N + colb] * ACC[r];   \
        }                                                                     \
      }                                                                       \
    }
    EPI_NT(acc0, 0) EPI_NT(acc1, 1)
#undef EPI_NT
  }
}

// ---- TN partial: part[seg] = A(R,M)^T @ B(R,N) over R-segment; N=512 ----
// rps is a multiple of 32; only the final segment may have a ragged tail.
__global__ __launch_bounds__(256) void gemm_tn_part(
    const float* __restrict__ A, const float* __restrict__ Bm,
    float* __restrict__ part, int R, int M, int N, int rps) {
  __shared__ __align__(16) float sAr[32 * 36];   // [r][m]
  __shared__ __align__(16) float sB[32 * 132];   // [r][n]
  const int tid = threadIdx.x;
  const int half = (tid >> 4) & 1, lrow = tid & 15;
  const int wave = tid >> 5;
  const int m_sub = wave & 1, n_base = (wave >> 1) * 2;
  const int mblk = blockIdx.x, nblk = blockIdx.y, seg = blockIdx.z;
  int r0s = seg * rps;
  int r1s = r0s + rps; if (r1s > R) r1s = R;
  v8f acc0 = {}, acc1 = {};
  for (int r0 = r0s; r0 < r1s; r0 += 32) {
    int rem = r1s - r0;
    if (rem >= 32) {  // block-uniform branch
      {  // 32x32 A chunk: exactly one b128 per lane
        int r = tid >> 3, c4 = (tid & 7) << 2;
        cp16_g2l(&A[(size_t)(r0 + r) * M + mblk * 32 + c4], &sAr[r * 36 + c4]);
      }
      for (int idx = tid; idx < 32 * 32; idx += 256) {
        int r = idx >> 5, c4 = (idx & 31) << 2;
        cp16_g2l(&Bm[(size_t)(r0 + r) * N + nblk * 128 + c4],
                 &sB[r * 132 + c4]);
      }
      async_wait0();
    } else {  // ragged tail: zero-fill rows >= rem
      for (int idx = tid; idx < 32 * 32; idx += 256) {
        int r = idx >> 5, c = idx & 31;
        sAr[r * 36 + c] =
            (r < rem) ? A[(size_t)(r0 + r) * M + mblk * 32 + c] : 0.0f;
      }
      for (int idx = tid; idx < 32 * 128; idx += 256) {
        int r = idx >> 7, c = idx & 127;
        sB[r * 132 + c] =
            (r < rem) ? Bm[(size_t)(r0 + r) * N + nblk * 128 + c] : 0.0f;
      }
    }
    __syncthreads();
    for (int kk = 0; kk < 32; kk += 4) {
      v2f a;
      a.x = sAr[(kk + 2 * half) * 36 + m_sub * 16 + lrow];
      a.y = sAr[(kk + 1 + 2 * half) * 36 + m_sub * 16 + lrow];
      const float* bp0 = &sB[(kk + 2 * half) * 132 + n_base * 16 + lrow];
      const float* bp1 = &sB[(kk + 1 + 2 * half) * 132 + n_base * 16 + lrow];
      v2f b;
      b.x = bp0[0];  b.y = bp1[0];  acc0 = wmma4(a, b, acc0);
      b.x = bp0[16]; b.y = bp1[16]; acc1 = wmma4(a, b, acc1);
    }
    __syncthreads();
  }
  int rowb = mblk * 32 + m_sub * 16;
  float* pseg = part + (size_t)seg * M * N;
#define EPI_TN(ACC, JJ)                                                       \
  {                                                                           \
    int colb = nblk * 128 + (n_base + (JJ)) * 16 + lrow;                      \
    for (int r = 0; r < 8; ++r) {                                             \
      int row = rowb + r + 8 * half;                                          \
      pseg[(size_t)row * N + colb] = ACC[r];                                  \
    }                                                                         \
  }
  EPI_TN(acc0, 0) EPI_TN(acc1, 1)
#undef EPI_TN
}

__global__ void tn_reduce(const float* __restrict__ part,
                          const float* __restrict__ theta,
                          float* __restrict__ xc, int MN, int segs) {
  int i = blockIdx.x * 256 + threadIdx.x;
  if (i < MN) {
    float s = 0.0f;
    for (int q = 0; q < segs; ++q) s += part[(size_t)q * MN + i];
    xc[i] = theta[i] * s;
  }
}

// ---- column sums: Sw, Sphi, Sn = sum sqrt(phi)*noise (stage 1) ----
__global__ void moments_k(const float* __restrict__ W,
                          const float* __restrict__ phi,
                          const float* __restrict__ noise,
                          float* __restrict__ pw, float* __restrict__ pp,
                          float* __restrict__ pn, int D, int K, int rps) {
  int k = threadIdx.x, seg = blockIdx.x;
  int r0 = seg * rps, r1 = r0 + rps; if (r1 > D) r1 = D;
  float sw = 0.0f, sp = 0.0f, sn = 0.0f;
  for (int r = r0; r < r1; ++r) {
    size_t i = (size_t)r * K + k;
    sw += W[i];
    float ph = phi[i];
    sp += ph;
    sn += sqrtf(ph) * noise[i];
  }
  pw[seg * K + k] = sw; pp[seg * K + k] = sp; pn[seg * K + k] = sn;
}

// ---- derive per-column alpha, beta, gamma from the TLASGR algebra ----
__global__ void scalars_k(const float* __restrict__ pw,
                          const float* __restrict__ pp,
                          const float* __restrict__ pn,
                          float* __restrict__ coeff, int K, int segs, float Df) {
  int k = threadIdx.x;
  float sw = 0.0f, sp = 0.0f, sn = 0.0f;
  for (int s = 0; s < segs; ++s) {
    sw += pw[s * K + k]; sp += pp[s * K + k]; sn += pn[s * K + k];
  }
  float ndot = 100.0f * sw;                        // MBRATIO * sum(WSZS)
  float inv_n = 1.0f / fmaxf(ndot, EPSC);
  float tmpsum = 100.0f * sw + 0.1f * Df;          // sum(EWSZS + 0.1)
  float gam = sqrtf(2.0f * inv_n);                 // EPSIT = 1
  float stepsum = sp * (1.0f - inv_n * tmpsum) + inv_n * tmpsum + gam * sn;
  float alpha = 2.0f - inv_n * tmpsum - stepsum;
  coeff[k] = alpha; coeff[K + k] = inv_n; coeff[2 * K + k] = gam;
}

// ---- p = max(EPS, alpha*phi + beta*(100W+0.1) + gamma*sqrt(phi)*noise) ----
__global__ void passb_k(const float* __restrict__ W,
                        const float* __restrict__ phi,
                        const float* __restrict__ noise,
                        const float* __restrict__ coeff,
                        float* __restrict__ outp, float* __restrict__ pps,
                        int D, int K, int rps) {
  int k = threadIdx.x, seg = blockIdx.x;
  float al = coeff[k], be = coeff[K + k], ga = coeff[2 * K + k];
  int r0 = seg * rps, r1 = r0 + rps; if (r1 > D) r1 = D;
  float s = 0.0f;
  for (int r = r0; r < r1; ++r) {
    size_t i = (size_t)r * K + k;
    float ph = phi[i];
    float p = al * ph + be * (100.0f * W[i] + 0.1f) + ga * sqrtf(ph) * noise[i];
    p = fmaxf(EPSC, p);
    outp[i] = p;
    s += p;
  }
  pps[seg * K + k] = s;
}

__global__ void psum_reduce_k(const float* __restrict__ pps,
                              float* __restrict__ inv, int K, int segs) {
  int k = threadIdx.x;
  float s = 0.0f;
  for (int q = 0; q < segs; ++q) s += pps[q * K + k];
  inv[k] = 1.0f / fmaxf(EPSC, s);
}

__global__ void norm_k(float* __restrict__ outp, const float* __restrict__ inv,
                       int n, int K) {
  int i = blockIdx.x * 256 + threadIdx.x;
  if (i < n) outp[i] *= inv[i & (K - 1)];  // K is a power of two
}

extern "C" void kernel_launch(void* const* d_in, const int* in_sizes, int n_in,
                              void* d_out, int out_size, void* d_ws,
                              size_t ws_size, hipStream_t stream) {
  (void)in_sizes; (void)n_in; (void)out_size; (void)ws_size;
  const int V = 30000, Bb = 512;
  const int Kt[3] = {128, 64, 32};
  const float* x = (const float*)d_in[0];
  const float* theta[3] = {(const float*)d_in[1], (const float*)d_in[2],
                           (const float*)d_in[3]};
  const float* phi[3] = {(const float*)d_in[4], (const float*)d_in[5],
                         (const float*)d_in[6]};
  const float* noise[3] = {(const float*)d_in[7], (const float*)d_in[8],
                           (const float*)d_in[9]};
  float* ws = (float*)d_ws;
  size_t o = 0;
  float* xT = ws + o;    o += (size_t)V * Bb;
  float* ratio = ws + o; o += (size_t)V * Bb;
  float* Wbuf = ws + o;  o += (size_t)V * 128;
  float* xc1 = ws + o;   o += 128 * 512;
  float* xc2 = ws + o;   o += 64 * 512;
  float* part = ws + o;  o += (size_t)8 * 128 * 512;
  float* pw = ws + o;    o += 128 * 128;
  float* pp = ws + o;    o += 128 * 128;
  float* pn = ws + o;    o += 128 * 128;
  float* coeff = ws + o; o += 3 * 128;
  float* pps = ws + o;   o += 128 * 128;
  float* invb = ws + o;  o += 128;
  float* outp = (float*)d_out;

  transpose_k<<<dim3((V + 31) / 32, (Bb + 31) / 32), 256, 0, stream>>>(x, xT,
                                                                       Bb, V);

  const int out_off[3] = {0, V * 128, V * 128 + 128 * 64};
  const float* xcin = xT;
  for (int t = 0; t < 3; ++t) {
    int D = (t == 0) ? V : Kt[t - 1];
    int K = Kt[t];
    // ratio_t = f(phi@theta, xc)
    gemm_nn_ratio<<<dim3((D + 63) / 64, Bb / 128), 256, 0, stream>>>(
        phi[t], theta[t], xcin, ratio, D, Bb, K, (t == 0) ? 0 : 1);
    // W_t = phi .* (ratio @ theta^T)
    gemm_nt_w<<<dim3((D + 63) / 64, (K + 63) / 64), 256, 0, stream>>>(
        ratio, theta[t], phi[t], Wbuf, D, K, Bb);
    // x_{t+1} = theta .* (phi^T @ ratio)   (not needed after layer 1)
    if (t < 2) {
      int segs = (t == 0) ? 8 : 1;
      int rps = (D + segs - 1) / segs;
      rps = (rps + 31) & ~31;  // multiple of 32: only final segment ragged
      float* xnext = (t == 0) ? xc1 : xc2;
      gemm_tn_part<<<dim3(K / 32, Bb / 128, segs), 256, 0, stream>>>(
          phi[t], ratio, part, D, K, Bb, rps);
      tn_reduce<<<dim3((K * Bb + 255) / 256), 256, 0, stream>>>(
          part, theta[t], xnext, K * Bb, segs);
      xcin = xnext;
    }
    // TLASGR + simplex projection + normalization
    int msegs = (t == 0) ? 128 : ((D + 31) / 32);
    int mrps = (D + msegs - 1) / msegs;
    moments_k<<<msegs, K, 0, stream>>>(Wbuf, phi[t], noise[t], pw, pp, pn, D, K,
                                       mrps);
    scalars_k<<<1, K, 0, stream>>>(pw, pp, pn, coeff, K, msegs, (float)D);
    passb_k<<<msegs, K, 0, stream>>>(Wbuf, phi[t], noise[t], coeff,
                                     outp + out_off[t], pps, D, K, mrps);
    psum_reduce_k<<<1, K, 0, stream>>>(pps, invb, K, msegs);
    norm_k<<<dim3((D * K + 255) / 256), 256, 0, stream>>>(outp + out_off[t],
                                                          invb, D * K, K);
  }
}